// SymmetryControl_48034914238616
// MI455X (gfx1250) — compile-verified
//
#include <hip/hip_runtime.h>
#include <cstdint>

// Problem geometry (fixed by the reference): B=16, C=96, H=96, W=96, step=12.
#define WDIM 96
#define ROWS_PER_BLOCK 8
#define CH 9216  // C*H, rows per batch image

#if defined(__has_builtin)
#if __has_builtin(__builtin_amdgcn_global_load_async_to_lds_b128) && \
    __has_builtin(__builtin_amdgcn_s_wait_asynccnt)
#define USE_ASYNC_LDS_B128 1
#elif __has_builtin(__builtin_amdgcn_global_load_async_to_lds_b32) && \
    __has_builtin(__builtin_amdgcn_s_wait_asynccnt)
#define USE_ASYNC_LDS_B32 1
#endif
#endif

typedef int v4i_ __attribute__((vector_size(16)));
typedef __attribute__((address_space(1))) int* gas_i32_t;
typedef __attribute__((address_space(3))) int* lds_i32_t;
typedef __attribute__((address_space(1))) v4i_* gas_v4i_t;
typedef __attribute__((address_space(3))) v4i_* lds_v4i_t;

__device__ __forceinline__ float sigmoidf_(float v) {
  return 1.0f / (1.0f + __expf(-v));
}

__global__ __launch_bounds__(WDIM * ROWS_PER_BLOCK)
void SymmetryControl_kernel(const float* __restrict__ x,
                            const float* __restrict__ s,
                            const float* __restrict__ w,
                            float* __restrict__ out,
                            int total_rows) {
  __shared__ __align__(16) float sx[ROWS_PER_BLOCK][WDIM];
  __shared__ __align__(16) float sw[ROWS_PER_BLOCK][WDIM];

  const int i = threadIdx.x;                 // angular index 0..95
  const int r = threadIdx.y;                 // row slot within block
  const int row = blockIdx.x * ROWS_PER_BLOCK + r;
  const bool live = row < total_rows;
  const int base = row * WDIM;               // < 2^31 elements total

#if defined(USE_ASYNC_LDS_B128)
  // Stage one 96-float row with 24 x 16B async DMAs: lanes 0..23 stage x,
  // lanes 24..47 stage w, lanes 48..95 skip (they still hit the barrier).
  if (live) {
    if (i < 24) {
      __builtin_amdgcn_global_load_async_to_lds_b128(
          (gas_v4i_t)(x + base + 4 * i), (lds_v4i_t)&sx[r][4 * i], 0, 0);
    } else if (i < 48) {
      const int c = i - 24;
      __builtin_amdgcn_global_load_async_to_lds_b128(
          (gas_v4i_t)(w + base + 4 * c), (lds_v4i_t)&sw[r][4 * c], 0, 0);
    }
  }
#elif defined(USE_ASYNC_LDS_B32)
  if (live) {
    __builtin_amdgcn_global_load_async_to_lds_b32(
        (gas_i32_t)(x + base + i), (lds_i32_t)&sx[r][i], 0, 0);
    __builtin_amdgcn_global_load_async_to_lds_b32(
        (gas_i32_t)(w + base + i), (lds_i32_t)&sw[r][i], 0, 0);
  }
#else
  if (live) {
    sx[r][i] = x[base + i];
    sw[r][i] = w[base + i];
  }
#endif

  // Per-batch sigmoid gates (s is [B,5], tiny -> L2 resident). Computed while
  // the async copies are in flight.
  const int b = row / CH;
  float sg0 = 0.f, sg1 = 0.f, sg2 = 0.f, sg3 = 0.f, sg4 = 0.f;
  if (live) {
    const float* sb = s + b * 5;
    sg0 = sigmoidf_(sb[0]);
    sg1 = sigmoidf_(sb[1]);
    sg2 = sigmoidf_(sb[2]);
    sg3 = sigmoidf_(sb[3]);
    sg4 = sigmoidf_(sb[4]);
  }

#if defined(USE_ASYNC_LDS_B128) || defined(USE_ASYNC_LDS_B32)
  __builtin_amdgcn_s_wait_asynccnt(0);  // this wave's DMAs landed in LDS
#endif
  __syncthreads();                      // make all rows visible block-wide

  if (!live) return;

  // roll indices: out[i] uses v[(i-n) mod 96] for n = 24,48,72
  int i90  = i + 72; if (i90  >= WDIM) i90  -= WDIM;
  int i180 = i + 48; if (i180 >= WDIM) i180 -= WDIM;
  int i270 = i + 24; if (i270 >= WDIM) i270 -= WDIM;
  // mirror indices: v[(23-i) mod 96] and v[(47-i) mod 96]
  int if90  = 119 - i; if (if90  >= WDIM) if90  -= WDIM;
  int if180 = 143 - i; if (if180 >= WDIM) if180 -= WDIM;

  const float* rx = sx[r];
  const float* rw = sw[r];

  const float x0 = rx[i];
  const float w0 = rw[i];

  const float wg90   = sg0 * rw[i90];
  const float wg180  = sg1 * rw[i180];
  const float wg270  = sg2 * rw[i270];
  const float wgf90  = sg3 * rw[if90];
  const float wgf180 = sg4 * rw[if180];

  float a = w0 * x0;
  a = fmaf(wg90,   rx[i90],   a);
  a = fmaf(wg180,  rx[i180],  a);
  a = fmaf(wg270,  rx[i270],  a);
  a = fmaf(wgf90,  rx[if90],  a);
  a = fmaf(wgf180, rx[if180], a);

  const float bden = w0 + wg90 + wg180 + wg270 + wgf90 + wgf180;

  out[base + i] = a / bden;
}

extern "C" void kernel_launch(void* const* d_in, const int* in_sizes, int n_in,
                              void* d_out, int out_size, void* d_ws, size_t ws_size,
                              hipStream_t stream) {
  const float* x = (const float*)d_in[0];  // [16,96,96,96]
  const float* s = (const float*)d_in[1];  // [16,5]
  const float* w = (const float*)d_in[2];  // [16,96,96,96]
  float* out = (float*)d_out;

  const int total_rows = in_sizes[0] / WDIM;  // B*C*H = 147456
  const int grid = (total_rows + ROWS_PER_BLOCK - 1) / ROWS_PER_BLOCK;
  dim3 block(WDIM, ROWS_PER_BLOCK);

  SymmetryControl_kernel<<<grid, block, 0, stream>>>(x, s, w, out, total_rows);
}